// CustomLayerPCEN_47132971107164
// MI455X (gfx1250) — compile-verified
//
#include <hip/hip_runtime.h>

// PCEN on [1024 bands x 32768 frames] f32.
//
// Roofline: 256 MB min traffic -> ~11 us @ 23.3 TB/s; this kernel is built to
// sit on that bound. EMA scan runs on the matrix pipe as a triangular-matrix
// WMMA (v_wmma_f32_16x16x4_f32): m_block = L*X_block + p*carry, with
// L[j][k]=0.5^(j-k+1). Time is chunked with a 64-frame halo: 0.5^65 is below
// one fp32 ulp of any partial sum, so chunks decouple exactly -> 4096
// independent waves, no cross-workgroup scan passes.
//
// Input path: global_load_async_to_lds_b128 (ASYNCcnt DMA) into ping-pong LDS
// tiles -- zero VGPR staging, next tile streams behind current tile's compute.

typedef __attribute__((ext_vector_type(2))) float v2f;
typedef __attribute__((ext_vector_type(8))) float v8f;

#define N_BANDS   1024
#define N_FRAMES  32768
#define CF        512               // output frames per wave-chunk
#define CHUNKS    (N_FRAMES / CF)   // 64
#define BTILES    (N_BANDS / 16)    // 64
#define WPB       4                 // waves per block
#define TSTRIDE   68                // LDS row stride (floats): 16B-aligned, conflict-free band reads
#define STG       64                // frames staged per LDS tile
#define PCEN_EPS  1e-6f

__device__ __forceinline__ float pow2neg(int n) {
  // exact 2^-n for 0 <= n <= 126
  unsigned bits = (unsigned)(127 - n) << 23;
  return __builtin_bit_cast(float, bits);
}

__device__ __forceinline__ float fast_log2(float v) { return __builtin_amdgcn_logf(v); }   // v_log_f32
__device__ __forceinline__ float fast_exp2(float v) { return __builtin_amdgcn_exp2f(v); }  // v_exp_f32

// Async DMA: 16B global -> LDS per lane, tracked by ASYNCcnt (no VGPR data).
__device__ __forceinline__ void async_load_b128(float* ldsDst, const float* gSrc) {
    // generic shared pointer: low 32 bits are the LDS byte address
    unsigned lds = (unsigned)(unsigned long long)ldsDst;
    asm volatile("global_load_async_to_lds_b128 %0, %1, off"
                 :: "v"(lds), "v"(gSrc) : "memory");
}
__device__ __forceinline__ void wait_async_le(int n) {
    if (n == 0) asm volatile("s_wait_asynccnt 0x0" ::: "memory");
    else        asm volatile("s_wait_asynccnt 0x8" ::: "memory");
}

__global__ __launch_bounds__(WPB * 32, 1)
void pcen_wmma_kernel(const float* __restrict__ x,
                      const float* __restrict__ pAlpha,
                      const float* __restrict__ pR,
                      const float* __restrict__ pDelta,
                      float* __restrict__ out)
{
    const int lane = threadIdx.x & 31;
    const int w    = threadIdx.x >> 5;
    const int wlin = blockIdx.x * WPB + w;

    const int bt = wlin / CHUNKS;       // band tile [0,64)
    const int ch = wlin % CHUNKS;       // time chunk [0,64)
    const int bandBase = bt * 16;

    const int hi   = lane >> 4;         // lane half
    const int band = lane & 15;         // = M for A, = N for B/C/D
    const int col4 = (lane & 15) << 2;  // float4 column for bulk ld/st

    const float alpha = pAlpha[0];
    const float r     = pR[0];
    const float delta = pDelta[0];
    const float deltar = fast_exp2(r * fast_log2(delta));

    // A = L in WMMA A layout: lane holds M=band, VGPR g holds K = g + 2*hi (per K=4 slice)
    v2f A[4];
#pragma unroll
    for (int s = 0; s < 4; ++s) {
        int k0 = 4 * s + 2 * hi;        // full-k of .x ; .y is k0+1
        // L[j][k] = 0.5^(j-k+1)
        A[s].x = (k0     <= band) ? pow2neg(band - k0 + 1) : 0.0f;
        A[s].y = (k0 + 1 <= band) ? pow2neg(band - k0)     : 0.0f;
    }

    // carry weights p[j] = 0.5^(j+1), j = g + 8*hi (C/D row per VGPR g)
    float pj[8];
#pragma unroll
    for (int g = 0; g < 8; ++g) pj[g] = pow2neg(g + 8 * hi + 1);

    // ping-pong LDS tiles per wave
    __shared__ __align__(16) float tile[WPB][2][16][TSTRIDE];

    const int outStart  = ch * CF;
    const int scanStart = (ch == 0) ? 0 : (outStart - STG);
    const int nStages   = (outStart + CF - scanStart) / STG;  // 8 or 9

    float carry = 0.0f;

    // prologue: async-fill tile 0 (8 x b128 per lane-group = 4KB tile)
#pragma unroll
    for (int rr = 0; rr < 8; ++rr) {
        const int row = 2 * rr + hi;
        async_load_b128(&tile[w][0][row][col4],
                        x + (size_t)(bandBase + row) * N_FRAMES + scanStart + col4);
    }

    for (int st = 0; st < nStages; ++st) {
        const int buf = st & 1;
        const int gf0 = scanStart + st * STG;
        float (*T)[TSTRIDE] = tile[w][buf];

        // stream next stage into the other buffer, then wait for current tile.
        // Async loads retire in order: <=8 outstanding => current 8 are done.
        const int more = __builtin_amdgcn_readfirstlane((st + 1 < nStages) ? 1 : 0);
        if (more) {
            const int gfn = gf0 + STG;
            float (*Tn)[TSTRIDE] = tile[w][buf ^ 1];
#pragma unroll
            for (int rr = 0; rr < 8; ++rr) {
                const int row = 2 * rr + hi;
                async_load_b128(&Tn[row][col4],
                                x + (size_t)(bandBase + row) * N_FRAMES + gfn + col4);
            }
            wait_async_le(8);
        } else {
            wait_async_le(0);
        }

        const int emit = __builtin_amdgcn_readfirstlane((gf0 >= outStart) ? 1 : 0);

        // 4 blocks of 16 frames; scan each via 4 chained f32 WMMAs
#pragma unroll
        for (int blk = 0; blk < 4; ++blk) {
            const int f0 = blk << 4;
            v8f acc = {0.f, 0.f, 0.f, 0.f, 0.f, 0.f, 0.f, 0.f};

#pragma unroll
            for (int s = 0; s < 4; ++s) {
                // B layout: lane holds N=band, VGPR g holds K = g + 2*hi
                const int fb = f0 + 4 * s + 2 * hi;
                v2f b;
                b.x = T[band][fb];
                b.y = T[band][fb + 1];
                acc = __builtin_amdgcn_wmma_f32_16x16x4_f32(
                    false, A[s], false, b, (short)0, acc, false, false);
            }

            // add carry term: m[j] += 0.5^(j+1) * carry[band]
#pragma unroll
            for (int g = 0; g < 8; ++g) acc[g] = fmaf(pj[g], carry, acc[g]);

            // new carry = m[15][band], held by lane 16+band in acc[7]
            carry = __shfl(acc[7], 16 + band, 32);

            if (emit) {
                // PCEN elementwise; C/D layout: VGPR g -> frame f0 + g + 8*hi, col band
#pragma unroll
                for (int g = 0; g < 8; ++g) {
                    const int f = f0 + g + 8 * hi;
                    const float xx = T[band][f];
                    const float m  = acc[g];
                    const float inva = fast_exp2(-alpha * fast_log2(PCEN_EPS + m));
                    const float v    = fmaf(xx, inva, delta);
                    const float res  = fast_exp2(r * fast_log2(v)) - deltar;
                    T[band][f] = res;   // in-place; this block's x is consumed
                }
            }
        }

        if (emit) {
            // coalesced bulk store of the finished 16x64 tile (256B per row-half)
#pragma unroll
            for (int rr = 0; rr < 8; ++rr) {
                const int row = 2 * rr + hi;
                const float4 v = *reinterpret_cast<const float4*>(&T[row][col4]);
                *reinterpret_cast<float4*>(
                    out + (size_t)(bandBase + row) * N_FRAMES + gf0 + col4) = v;
            }
        }
    }
}

extern "C" void kernel_launch(void* const* d_in, const int* in_sizes, int n_in,
                              void* d_out, int out_size, void* d_ws, size_t ws_size,
                              hipStream_t stream) {
    const float* x      = (const float*)d_in[0];
    const float* alpha  = (const float*)d_in[1];
    const float* r      = (const float*)d_in[2];
    const float* delta  = (const float*)d_in[3];
    float* out = (float*)d_out;

    const int totalWaves = BTILES * CHUNKS;        // 4096
    const int blocks = totalWaves / WPB;           // 1024
    pcen_wmma_kernel<<<blocks, WPB * 32, 0, stream>>>(x, alpha, r, delta, out);
}